// STGCNLayer_73924977099264
// MI455X (gfx1250) — compile-verified
//
#include <hip/hip_runtime.h>

typedef __attribute__((ext_vector_type(2))) float v2f;
typedef __attribute__((ext_vector_type(8))) float v8f;

#define C_IN   64
#define C_OUT  64
#define T_DIM  4
#define KT_W   3
#define CT     (C_OUT * T_DIM)   // 256 floats per node
#define XS_ROW    66             // LDS floats per time-row (64 + 2 bank skew)
#define XS_NSTRIDE 396           // LDS floats per node: 6*66 (mod 64 = 12)
#define BG_PAIRS  (16 * 2 * 64)  // spatial  B pairs: s(16) x h(2) x o(64)
#define BT_PAIRS  (48 * 2 * 64)  // temporal B pairs: kk*16+s(48) x h(2) x o(64)

// ---------------------------------------------------------------- helpers
__global__ void zero_f32_kernel(float* __restrict__ p, int n) {
  int i = blockIdx.x * blockDim.x + threadIdx.x;
  if (i < n) p[i] = 0.0f;
}

__global__ void degree_kernel(const long long* __restrict__ dst,
                              float* __restrict__ deg, int E) {
  int e = blockIdx.x * blockDim.x + threadIdx.x;
  if (e < E) atomicAdd(&deg[(int)dst[e]], 1.0f);
}

__global__ void dinv_kernel(float* __restrict__ deg, int N) {
  int i = blockIdx.x * blockDim.x + threadIdx.x;
  if (i < N) deg[i] = rsqrtf(deg[i] + 1.0f);
}

// Pack W_g / W_t into pair-interleaved B-fragment tables so each WMMA's
// B operand is a single aligned b64 load.
__global__ void pack_weights_kernel(const float* __restrict__ Wg,
                                    const float* __restrict__ Wt,
                                    float* __restrict__ Bg,
                                    float* __restrict__ Bt) {
  const int i = blockIdx.x * blockDim.x + threadIdx.x;
  if (i < BG_PAIRS) {
    const int o = i & 63, hh = (i >> 6) & 1, s = i >> 7;
    Bg[2 * i + 0] = Wg[(4 * s + 0 + hh) * C_OUT + o];
    Bg[2 * i + 1] = Wg[(4 * s + 2 + hh) * C_OUT + o];
  }
  if (i < BT_PAIRS) {
    const int o = i & 63, hh = (i >> 6) & 1, s = (i >> 7) & 15, kk = i >> 11;
    Bt[2 * i + 0] = Wt[(size_t)o * (C_IN * KT_W) + (4 * s + 0 + hh) * KT_W + kk];
    Bt[2 * i + 1] = Wt[(size_t)o * (C_IN * KT_W) + (4 * s + 2 + hh) * KT_W + kk];
  }
}

// ---------------------------------------------------------------- fused GEMM
// Block = 8 waves = 2 row-groups x 4 col-tiles; row-group = 16 rows m=n*T+t
// (4 nodes). x staged in LDS transposed+padded: xs[node][(t+1)*66 + c],
// rows t=-1 and t=4 zeroed -> every A fragment is one ds_load_b64 and the
// temporal im2col taps are plain row shifts. 64 V_WMMA_F32_16X16X4_F32.
__global__ __launch_bounds__(256)
void stgcn_gemm_kernel(const float* __restrict__ x,
                       const float* __restrict__ Bg,   // packed spatial B
                       const float* __restrict__ Bt,   // packed temporal B
                       const float* __restrict__ bg,
                       const float* __restrict__ bt,
                       const float* __restrict__ dinv,
                       float* __restrict__ h,
                       float* __restrict__ out,
                       int M)                           // M = N*T, mult. of 16
{
  __shared__ __attribute__((aligned(16))) float xs[2][4 * XS_NSTRIDE];

  const int tid  = threadIdx.x;
  const int wave = tid >> 5;
  const int lane = tid & 31;
  const int rgLocal = wave >> 2;                 // row-group within block
  const int colTile = wave & 3;
  const int m0 = (blockIdx.x * 2 + rgLocal) * 16;
  const int o0 = colTile * 16;

  // ---- stage x: threads 0-127 -> group 0, 128-255 -> group 1.
  // Each thread: 2 (node,c) pairs; coalesced b128 global load, 6 ds_store_b32
  // into the transposed rows (incl. the two zero boundary rows).
  {
    const int g   = tid >> 7;
    const int t2  = tid & 127;
    const int gm0 = (blockIdx.x * 2 + g) * 16;
    if (gm0 < M) {
      const int gnode0 = gm0 >> 2;
#pragma unroll
      for (int r = 0; r < 2; ++r) {
        const int p    = t2 + r * 128;           // (node,c) pair 0..255
        const int nloc = p >> 6;
        const int c    = p & 63;
        const float4 v = *(const float4*)(
            x + (size_t)(gnode0 + nloc) * (C_IN * T_DIM) + c * T_DIM);
        float* dst = &xs[g][nloc * XS_NSTRIDE + c];
        dst[0 * XS_ROW] = 0.0f;
        dst[1 * XS_ROW] = v.x;
        dst[2 * XS_ROW] = v.y;
        dst[3 * XS_ROW] = v.z;
        dst[4 * XS_ROW] = v.w;
        dst[5 * XS_ROW] = 0.0f;
      }
    }
  }
  __syncthreads();

  if (m0 >= M) return;                           // wave-uniform

  const int halfSel = lane >> 4;                 // 0: lanes 0-15, 1: 16-31
  const int l16     = lane & 15;
  const int nloc    = l16 >> 2;                  // node within row-group
  const int tA      = l16 & 3;                   // time of this A-row
  const float* __restrict__ xbase = xs[rgLocal] + nloc * XS_NSTRIDE;
  const int oB  = o0 + l16;
  const int cA0 = halfSel * 2;                   // lane's first A channel

  const v2f* __restrict__ BgP = (const v2f*)Bg;
  const v2f* __restrict__ BtP = (const v2f*)Bt;

  // ---- spatial GEMM (K = 64)
  const float* __restrict__ xrowS = xbase + (tA + 1) * XS_ROW + cA0;
  v8f acc_h = {};
#pragma unroll
  for (int s = 0; s < C_IN / 4; ++s) {
    const v2f a = *(const v2f*)(xrowS + s * 4);
    const v2f b = BgP[(s * 2 + halfSel) * 64 + oB];
    acc_h = __builtin_amdgcn_wmma_f32_16x16x4_f32(
        false, a, false, b, (short)0, acc_h, false, false);
  }

  // ---- temporal GEMM (K = 192, permuted k = kk*64 + c):
  // A[m, kk*64+c] = x[n,c,tA+kk-1] = xbase[(tA+kk)*66 + c] (pad rows are 0)
  v8f acc_t = {};
#pragma unroll
  for (int kk = 0; kk < KT_W; ++kk) {
    const float* __restrict__ xrowT = xbase + (tA + kk) * XS_ROW + cA0;
#pragma unroll
    for (int s = 0; s < C_IN / 4; ++s) {
      const v2f a = *(const v2f*)(xrowT + s * 4);
      const v2f b = BtP[((kk * 16 + s) * 2 + halfSel) * 64 + oB];
      acc_t = __builtin_amdgcn_wmma_f32_16x16x4_f32(
          false, a, false, b, (short)0, acc_t, false, false);
    }
  }

  // ---- epilogue. D layout: VGPR i -> row i+8*halfSel, col l16. Each half
  // of the accumulators is one node with t=0..3 contiguous -> float4 stores,
  // contiguous 256B across the 16-lane half.
  const float biases = bg[oB] + bt[oB];
#pragma unroll
  for (int half = 0; half < 2; ++half) {
    const int m = m0 + halfSel * 8 + half * 4;   // t = 0..3 within this node
    const int n = m >> 2;
    const float di  = dinv[n];
    const float di2 = di * di;
    float4 hv, ov;
    hv.x = acc_h[half * 4 + 0]; hv.y = acc_h[half * 4 + 1];
    hv.z = acc_h[half * 4 + 2]; hv.w = acc_h[half * 4 + 3];
    ov.x = acc_t[half * 4 + 0] + hv.x * di2 + biases;
    ov.y = acc_t[half * 4 + 1] + hv.y * di2 + biases;
    ov.z = acc_t[half * 4 + 2] + hv.z * di2 + biases;
    ov.w = acc_t[half * 4 + 3] + hv.w * di2 + biases;
    const size_t idx = (size_t)n * CT + (size_t)oB * T_DIM;
    *(float4*)(h + idx)   = hv;
    *(float4*)(out + idx) = ov;
  }
}

// ---------------------------------------------------------------- edge scatter
// One wave per edge: out[dst] += h[src] * dinv[src]*dinv[dst]  (256 floats,
// coalesced lanes; h (51MB) lives in the 192MB L2 so gathers mostly hit L2)
__global__ __launch_bounds__(256)
void edge_scatter_kernel(const long long* __restrict__ ei,
                         const float* __restrict__ h,
                         const float* __restrict__ dinv,
                         float* __restrict__ out, int E) {
  const int e = (int)((blockIdx.x * blockDim.x + threadIdx.x) >> 5);
  const int lane = threadIdx.x & 31;
  if (e >= E) return;
  const int s = (int)ei[e];
  const int d = (int)ei[(size_t)E + e];
  const float w = dinv[s] * dinv[d];
  const float* __restrict__ hs = h + (size_t)s * CT;
  float* __restrict__ od = out + (size_t)d * CT;
#pragma unroll
  for (int j = 0; j < CT / 32; ++j) {
    const int idx = lane + j * 32;
    atomicAdd(&od[idx], hs[idx] * w);
  }
}

// ---------------------------------------------------------------- launch
extern "C" void kernel_launch(void* const* d_in, const int* in_sizes, int n_in,
                              void* d_out, int out_size, void* d_ws, size_t ws_size,
                              hipStream_t stream) {
  const float*     x  = (const float*)d_in[0];
  const long long* ei = (const long long*)d_in[1];   // int64 (2, E)
  const float*     Wg = (const float*)d_in[2];
  const float*     bg = (const float*)d_in[3];
  const float*     Wt = (const float*)d_in[4];
  const float*     bt = (const float*)d_in[5];
  float* out = (float*)d_out;

  const int N = in_sizes[0] / (C_IN * T_DIM);
  const int E = in_sizes[1] / 2;
  const int M = N * T_DIM;

  // workspace: [deg/dinv : N pad] [h : N*256] [Bg : 4096] [Bt : 12288]
  float* deg = (float*)d_ws;
  float* h   = deg + (((size_t)N + 63) & ~(size_t)63);
  float* Bg  = h + (size_t)N * CT;
  float* Bt  = Bg + 2 * BG_PAIRS;

  zero_f32_kernel<<<(N + 255) / 256, 256, 0, stream>>>(deg, N);
  degree_kernel<<<(E + 255) / 256, 256, 0, stream>>>(ei + E, deg, E);
  dinv_kernel<<<(N + 255) / 256, 256, 0, stream>>>(deg, N);
  pack_weights_kernel<<<(BT_PAIRS + 255) / 256, 256, 0, stream>>>(Wg, Wt, Bg, Bt);

  const int rowTiles = M / 16;                  // exact: N*T multiple of 16
  const int blocks   = (rowTiles + 1) / 2;      // 2 row-groups per block
  stgcn_gemm_kernel<<<blocks, 256, 0, stream>>>(x, Bg, Bt, bg, bt, deg, h, out, M);

  const long long scatterThreads = (long long)E * 32;
  edge_scatter_kernel<<<(int)((scatterThreads + 255) / 256), 256, 0, stream>>>(
      ei, h, deg, out, E);
}